// MLPConstructor2_7172595384464
// MI455X (gfx1250) — compile-verified
//
#include <hip/hip_runtime.h>

// Problem constants (from the reference): B=8, N=2048, T=512, D=32
#define NB   8
#define NSP  2048
#define NTM  512
#define SS   2560            // N + T
#define DD   32
#define PLANE (NB * SS)      // 20480 floats per precomputed array

typedef __attribute__((ext_vector_type(2))) float v2f;
typedef __attribute__((ext_vector_type(4))) float v4f;
typedef __attribute__((ext_vector_type(8))) float v8f;

#if __has_builtin(__builtin_amdgcn_tanhf)
#define TANHF(x) __builtin_amdgcn_tanhf(x)
#else
#define TANHF(x) tanhf(x)
#endif

// -----------------------------------------------------------------------------
// Phase 1: projections via V_WMMA_F32_16X16X4_F32.
// One wave (32 lanes) per 16 node rows. A = 16x32 tile of node features,
// B columns 0..3 = the four length-32 weight vectors this node type needs.
// Chain 8 WMMAs over K. Result D[m][n] = dot(x_m, w_n), n in 0..3:
//   n=0 -> rowL (+biasL), n=1 -> rowR (+biasR), n=2 -> colT, n=3 -> colB.
// ws layout: [rowL | rowR | colT | colB], each PLANE floats, indexed [b*SS + i].
// -----------------------------------------------------------------------------
__global__ __launch_bounds__(32)
void mlp_adj_phase1(const float* __restrict__ sp, const float* __restrict__ tm,
                    const float* __restrict__ wss, const float* __restrict__ wtt,
                    const float* __restrict__ wst, const float* __restrict__ wts,
                    const float* __restrict__ bss, const float* __restrict__ btt,
                    const float* __restrict__ bst, const float* __restrict__ bts,
                    float* __restrict__ ws)
{
    const int chunk = blockIdx.x;             // 0 .. NB*(SS/16)-1
    const int b     = chunk / (SS / 16);
    const int i0    = (chunk % (SS / 16)) * 16;
    const int lane  = threadIdx.x;            // 0..31 (wave32)
    const int n     = lane & 15;              // matrix column / M-row index
    const int half  = lane >> 4;              // K half (A/B) and M half (C/D)

    const bool isSp = (i0 < NSP);             // N,T multiples of 16 -> no mixed chunk
    const float* src = isSp ? (sp + ((size_t)b * NSP + i0) * DD)
                            : (tm + ((size_t)b * NTM + (i0 - NSP)) * DD);
    // Weight vector per output column (see mapping in header comment)
    const float* wv0 = isSp ? wss        : wts;          // rowL: w[:D]
    const float* wv1 = isSp ? wst        : wtt;          // rowR: w[:D]
    const float* wv2 = isSp ? (wss + DD) : (wst + DD);   // colT: w[D:]
    const float* wv3 = isSp ? (wts + DD) : (wtt + DD);   // colB: w[D:]
    const float biasL = isSp ? bss[0] : bts[0];
    const float biasR = isSp ? bst[0] : btt[0];

    const bool  valid = (n < 4);
    const float* wp = (n == 0) ? wv0 : (n == 1) ? wv1 : (n == 2) ? wv2 : wv3;
    const float* arow = src + (size_t)n * DD;   // A row m = lane&15

    v8f c = {0.f, 0.f, 0.f, 0.f, 0.f, 0.f, 0.f, 0.f};
#pragma unroll
    for (int k = 0; k < 8; ++k) {
        // A 16x4 f32 layout: lanes 0-15 {K=0,1}, lanes 16-31 {K=2,3}
        v2f a;
        a.x = arow[4 * k + 2 * half + 0];
        a.y = arow[4 * k + 2 * half + 1];
        // B 4x16 f32: lane carries column n, VGPR/lane-half carries K
        float w0 = wp[4 * k + 2 * half + 0];
        float w1 = wp[4 * k + 2 * half + 1];
        v2f bm;
        bm.x = valid ? w0 : 0.f;
        bm.y = valid ? w1 : 0.f;
        c = __builtin_amdgcn_wmma_f32_16x16x4_f32(
                /*neg_a=*/false, a, /*neg_b=*/false, bm,
                /*c_mod=*/(short)0, c, /*reuse_a=*/false, /*reuse_b=*/false);
    }

    if (valid) {
        const float bias = (n == 0) ? biasL : (n == 1) ? biasR : 0.f;
        // C/D layout: lane holds D[M = r + 8*half][N = n] in c[r]
        float* dst = ws + (size_t)n * PLANE + (size_t)b * SS + i0 + 8 * half;
#pragma unroll
        for (int r = 0; r < 8; ++r) dst[r] = c[r] + bias;
    }
}

// -----------------------------------------------------------------------------
// Phase 2: the bandwidth kernel. 13.1M float4 stores (209.7 MB).
// Each lane: 1 cached scalar row-term load + 1 coalesced b128 col-term load,
// v_add + v_max + v_tanh per element, then a non-temporal global_store_b128.
// -----------------------------------------------------------------------------
__global__ __launch_bounds__(256)
void mlp_adj_phase2(const float* __restrict__ ws, float* __restrict__ out)
{
    const unsigned J4 = SS / 4;                          // 640 float4 per row
    unsigned flat = blockIdx.x * 256u + threadIdx.x;     // < NB*SS*J4 = 13,107,200
    unsigned j4  = flat % J4;
    unsigned t   = flat / J4;
    unsigned row = t % SS;
    unsigned b   = t / SS;
    if (b >= NB) return;

    const unsigned j = j4 * 4u;                          // N%4==0 -> no straddle
    const float r = ws[(j   < NSP ? 0u : (unsigned)PLANE)      + b * SS + row];
    const float* cp = ws + (row < NSP ? 2u * (unsigned)PLANE
                                      : 3u * (unsigned)PLANE) + b * SS + j;
    v4f cv = *(const v4f*)cp;

    v4f o;
    o.x = TANHF(fmaxf(r + cv.x, 0.f));   // tanh(relu(x)): tanh(0)=0
    o.y = TANHF(fmaxf(r + cv.y, 0.f));
    o.z = TANHF(fmaxf(r + cv.z, 0.f));
    o.w = TANHF(fmaxf(r + cv.w, 0.f));

    v4f* op = (v4f*)(out + ((size_t)b * SS + row) * SS + j);
    __builtin_nontemporal_store(o, op);  // streaming store: output is write-once
}

extern "C" void kernel_launch(void* const* d_in, const int* in_sizes, int n_in,
                              void* d_out, int out_size, void* d_ws, size_t ws_size,
                              hipStream_t stream)
{
    // setup_inputs order: spatial, temporal, then (w,b) for ss, tt, st, ts
    const float* sp  = (const float*)d_in[0];
    const float* tm  = (const float*)d_in[1];
    const float* wss = (const float*)d_in[2];
    const float* bss = (const float*)d_in[3];
    const float* wtt = (const float*)d_in[4];
    const float* btt = (const float*)d_in[5];
    const float* wst = (const float*)d_in[6];
    const float* bst = (const float*)d_in[7];
    const float* wts = (const float*)d_in[8];
    const float* bts = (const float*)d_in[9];

    float* ws  = (float*)d_ws;      // needs 4*PLANE*4 = 320 KB
    float* out = (float*)d_out;

    // Phase 1: B * (SS/16) waves, one wave per workgroup
    mlp_adj_phase1<<<NB * (SS / 16), 32, 0, stream>>>(
        sp, tm, wss, wtt, wst, wts, bss, btt, bst, bts, ws);

    // Phase 2: NB*SS*(SS/4) float4 = 13,107,200 threads
    const unsigned total  = (unsigned)NB * SS * (SS / 4);
    const unsigned blocks = (total + 255u) / 256u;       // = 51200 exactly
    mlp_adj_phase2<<<blocks, 256, 0, stream>>>(ws, out);
}